// MaskPatches_74818330296837
// MI455X (gfx1250) — compile-verified
//
#include <hip/hip_runtime.h>
#include <stdint.h>

// MaskPatches: out[c,t,y,x] = frames[c,t,y,x] * (patch (y/14, x/14) of frame t masked ? 0 : 1)
// Bandwidth-bound (0.125 FLOP/byte, ~308 MB @ 23.3 TB/s => ~13 us floor).
// CDNA5 paths: async global->LDS bitmask staging (ASYNCcnt), NT-hinted b128 streaming,
// global_prefetch_b8. No WMMA: no reduction axis exists for this op.

#define PATCH 14
static constexpr int CH   = 3;
static constexpr int TT   = 256;
static constexpr int HH   = 224;
static constexpr int WW   = 224;
static constexpr int SIDE = 16;          // 224/14
static constexpr int MI   = 128;         // masked patches per frame
static constexpr int PLANE  = HH * WW;   // 50176 floats per (c,t) plane
static constexpr int PLANE4 = PLANE / 4; // 12544 float4 per plane
static constexpr int SPLIT  = 7;         // 12544/7 = 1792 = 7*256
static constexpr int CHUNK4 = PLANE4 / SPLIT;
static constexpr int WPT    = (SIDE * SIDE) / 32; // 8 bitmask words per frame

typedef float f4 __attribute__((ext_vector_type(4)));
typedef __attribute__((address_space(3))) int* lds_i32p;
typedef __attribute__((address_space(1))) int* glb_i32p;

__global__ void MaskPatches_zero_kernel(uint32_t* __restrict__ bits) {
    int i = blockIdx.x * blockDim.x + threadIdx.x;
    if (i < TT * WPT) bits[i] = 0u;
}

__global__ void MaskPatches_scatter_kernel(const int* __restrict__ midx,
                                           uint32_t* __restrict__ bits) {
    int i = blockIdx.x * blockDim.x + threadIdx.x;
    if (i >= TT * MI) return;
    int t = i / MI;
    int v = midx[i];
    int h = v % SIDE;            // patch row (reference: idx % sidelen)
    int w = v / SIDE;            // patch col (reference: idx // sidelen)
    int p = h * SIDE + w;        // flat patch id in mask[t, h, w]
    atomicOr(&bits[t * WPT + (p >> 5)], 1u << (p & 31));
}

__global__ __launch_bounds__(256) void MaskPatches_apply_kernel(
        const float* __restrict__ in,
        const uint32_t* __restrict__ bits,
        float* __restrict__ out) {
    __shared__ uint32_t smask[WPT];

    int b     = blockIdx.x;
    int plane = b / SPLIT;       // c*TT + t ; planes are contiguous in [C,T,H,W]
    int chunk = b % SPLIT;
    int t     = plane % TT;

#if __has_builtin(__builtin_amdgcn_global_load_async_to_lds_b32)
    // gfx1250 async copy: global bitmask words -> LDS, tracked by ASYNCcnt.
    if (threadIdx.x < WPT) {
        glb_i32p src = (glb_i32p)(uintptr_t)(bits + t * WPT + threadIdx.x);
        lds_i32p dst = (lds_i32p)(uint32_t)(uintptr_t)(&smask[threadIdx.x]);
        __builtin_amdgcn_global_load_async_to_lds_b32(src, dst, 0, 0);
    }
    asm volatile("s_wait_asynccnt 0x0" ::: "memory");
#else
    if (threadIdx.x < WPT) smask[threadIdx.x] = bits[t * WPT + threadIdx.x];
#endif
    __syncthreads();

    long base = (long)plane * PLANE;
    const f4* __restrict__ in4  = (const f4*)(in  + base);
    f4*       __restrict__ out4 = (f4*)      (out + base);

    int iend = (chunk + 1) * CHUNK4;
    for (int i = chunk * CHUNK4 + (int)threadIdx.x; i < iend; i += 256) {
        // keep the HBM read stream primed (gfx1250 global_prefetch_b8)
        int ipf = i + 256;
        if (ipf >= PLANE4) ipf = PLANE4 - 1;
        __builtin_prefetch(&in4[ipf], 0, 1);

        int x4 = i % (WW / 4);                 // 0..55
        int y  = i / (WW / 4);                 // 0..223
        int x  = x4 * 4;

        int pr = y / PATCH;                    // patch row 0..15
        uint32_t word = smask[pr >> 1];        // ds_load_b32
        int sh = (pr & 1) * SIDE;

        int pc0 = x / PATCH;                   // patch col of first pixel
        int r   = x - pc0 * PATCH;             // 4 px straddle <= 2 patches
        int pc1 = pc0 + ((r + 1) >= PATCH ? 1 : 0);
        int pc2 = pc0 + ((r + 2) >= PATCH ? 1 : 0);
        int pc3 = pc0 + ((r + 3) >= PATCH ? 1 : 0);

        f4 v = __builtin_nontemporal_load(&in4[i]);   // b128, TH_NT (read-once stream)
        v.x = ((word >> (sh + pc0)) & 1u) ? 0.0f : v.x;
        v.y = ((word >> (sh + pc1)) & 1u) ? 0.0f : v.y;
        v.z = ((word >> (sh + pc2)) & 1u) ? 0.0f : v.z;
        v.w = ((word >> (sh + pc3)) & 1u) ? 0.0f : v.w;
        __builtin_nontemporal_store(v, &out4[i]);     // b128, TH_NT (write-once stream)
    }
}

extern "C" void kernel_launch(void* const* d_in, const int* in_sizes, int n_in,
                              void* d_out, int out_size, void* d_ws, size_t ws_size,
                              hipStream_t stream) {
    const float* frames = (const float*)d_in[0];     // [3,256,224,224] fp32
    const int*   midx   = (const int*)d_in[1];       // [256,128] int32
    float*       out    = (float*)d_out;
    uint32_t*    bits   = (uint32_t*)d_ws;           // 256*8 words = 8 KB

    MaskPatches_zero_kernel<<<(TT * WPT + 255) / 256, 256, 0, stream>>>(bits);
    MaskPatches_scatter_kernel<<<(TT * MI + 255) / 256, 256, 0, stream>>>(midx, bits);
    MaskPatches_apply_kernel<<<CH * TT * SPLIT, 256, 0, stream>>>(frames, bits, out);
}